// HybridQuantumBinaryClassifier_65481071406628
// MI455X (gfx1250) — compile-verified
//
#include <hip/hip_runtime.h>
#include <math.h>

typedef __attribute__((ext_vector_type(2))) float v2f;
typedef __attribute__((ext_vector_type(8))) float v8f;

#define IMGS_PER_BLOCK 16
#define THREADS 256

// Fused pipeline:  mean -> convq(2x2,s2) -> conv1(5x5,s2,p1)+relu -> pool2x2s1
//                  -> conv2(3x3,s2,p1)+relu -> pool2x2s1 -> flatten
//                  -> [WMMA] z = h@wp^T -> [WMMA] RBF vs support -> logit -> sigmoid
__global__ __launch_bounds__(THREADS)
void hqbc_fused_kernel(const float* __restrict__ x,
                       const float* __restrict__ wq, const float* __restrict__ bq,
                       const float* __restrict__ w1, const float* __restrict__ b1,
                       const float* __restrict__ w2, const float* __restrict__ b2,
                       const float* __restrict__ wp, const float* __restrict__ bp,
                       const float* __restrict__ support,
                       const float* __restrict__ wl, const float* __restrict__ bl,
                       float* __restrict__ out, int B)
{
    // 64000 bytes of LDS, regions aliased across stages:
    //   [0      .. 12544) Q    : convq out, 16 img * 4ch * 14*14
    //   [12544  .. 16000) C1   : conv1 out, 16 img * 6ch * 6*6     (peak = 64000 B)
    //   [0      ..  2400) P1   : pool1 out, 16 img * 6ch * 5*5     (aliases dead Q)
    //   [2400   ..  4560) C2   : conv2 out, 16 img * 15ch * 3*3
    //   [4560   ..  5520) FLAT : 16 img * 60
    //   [5520   ..  5584) Z    : 16 img * 4
    //   [5584   ..  5600) ZZ   : 16 img, ||z||^2
    //   [5600   ..  6560) WPB  : wp zero-padded to 16 cols: WPB[n*60+k]
    __shared__ float lds[16000];
    float* Q    = lds;
    float* C1   = lds + 12544;
    float* P1   = lds;
    float* C2   = lds + 2400;
    float* FLAT = lds + 4560;
    float* Z    = lds + 5520;
    float* ZZ   = lds + 5584;
    float* WPB  = lds + 5600;

    const int tid  = threadIdx.x;
    const int img0 = blockIdx.x * IMGS_PER_BLOCK;

    // ---- Stage 1+2 fused: channel mean + convq (x is read exactly once) ----
    for (int i = tid; i < IMGS_PER_BLOCK * 196; i += THREADS) {
        const int img = i / 196, p = i % 196;
        const int oy = p / 14, ox = p % 14;
        const int gimg = img0 + img;
        float m[2][2];
        if (gimg < B) {
            const float* xb = x + (size_t)gimg * 2352;
#pragma unroll
            for (int dy = 0; dy < 2; ++dy) {
                const int o = (2 * oy + dy) * 28 + 2 * ox;   // even -> 8B aligned
                float2 a0 = *(const float2*)(xb + o);
                float2 a1 = *(const float2*)(xb + 784 + o);
                float2 a2 = *(const float2*)(xb + 1568 + o);
                m[dy][0] = (a0.x + a1.x + a2.x) * (1.0f / 3.0f);
                m[dy][1] = (a0.y + a1.y + a2.y) * (1.0f / 3.0f);
            }
        } else {
            m[0][0] = m[0][1] = m[1][0] = m[1][1] = 0.0f;
        }
#pragma unroll
        for (int oc = 0; oc < 4; ++oc) {
            float v = bq[oc] + m[0][0] * wq[oc * 4 + 0] + m[0][1] * wq[oc * 4 + 1]
                             + m[1][0] * wq[oc * 4 + 2] + m[1][1] * wq[oc * 4 + 3];
            Q[(img * 4 + oc) * 196 + p] = v;
        }
    }
    __syncthreads();

    // ---- Stage 3: conv1 5x5 s2 p1 (+ReLU; commutes with the max-pool) ----
    for (int i = tid; i < IMGS_PER_BLOCK * 216; i += THREADS) {
        const int img = i / 216, r = i % 216;
        const int oc = r / 36, p = r % 36;
        const int oy = p / 6, ox = p % 6;
        float acc = b1[oc];
        for (int ic = 0; ic < 4; ++ic) {
            const float* qin = Q + (img * 4 + ic) * 196;
            const float* wv  = w1 + (oc * 4 + ic) * 25;
#pragma unroll
            for (int ky = 0; ky < 5; ++ky) {
                const int iy = oy * 2 - 1 + ky;
                if (iy < 0 || iy >= 14) continue;
#pragma unroll
                for (int kx = 0; kx < 5; ++kx) {
                    const int ix = ox * 2 - 1 + kx;
                    if (ix < 0 || ix >= 14) continue;
                    acc += qin[iy * 14 + ix] * wv[ky * 5 + kx];
                }
            }
        }
        C1[i] = fmaxf(acc, 0.0f);
    }
    __syncthreads();

    // ---- Stage 4: maxpool 2x2 stride 1 -> [6,5,5]; also stage wp B-panel ----
    for (int i = tid; i < IMGS_PER_BLOCK * 150; i += THREADS) {
        const int img = i / 150, r = i % 150;
        const int oc = r / 25, p = r % 25;
        const int oy = p / 5, ox = p % 5;
        const float* c = C1 + (img * 6 + oc) * 36;
        P1[i] = fmaxf(fmaxf(c[oy * 6 + ox],       c[oy * 6 + ox + 1]),
                      fmaxf(c[(oy + 1) * 6 + ox], c[(oy + 1) * 6 + ox + 1]));
    }
    // wp is [4,60] row-major; pad to a 16x60 B-panel so the WMMA B-operand
    // loads are unconditional ds_load_b64 (no per-step EXEC divergence).
    for (int i = tid; i < 960; i += THREADS)
        WPB[i] = (i < 240) ? wp[i] : 0.0f;
    __syncthreads();

    // ---- Stage 5: conv2 3x3 s2 p1 (+ReLU) -> [15,3,3] ----
    for (int i = tid; i < IMGS_PER_BLOCK * 135; i += THREADS) {
        const int img = i / 135, r = i % 135;
        const int oc = r / 9, p = r % 9;
        const int oy = p / 3, ox = p % 3;
        float acc = b2[oc];
        for (int ic = 0; ic < 6; ++ic) {
            const float* pin = P1 + (img * 6 + ic) * 25;
            const float* wv  = w2 + (oc * 6 + ic) * 9;
#pragma unroll
            for (int ky = 0; ky < 3; ++ky) {
                const int iy = oy * 2 - 1 + ky;
                if (iy < 0 || iy >= 5) continue;
#pragma unroll
                for (int kx = 0; kx < 3; ++kx) {
                    const int ix = ox * 2 - 1 + kx;
                    if (ix < 0 || ix >= 5) continue;
                    acc += pin[iy * 5 + ix] * wv[ky * 3 + kx];
                }
            }
        }
        C2[i] = fmaxf(acc, 0.0f);
    }
    __syncthreads();

    // ---- Stage 6: maxpool 2x2 s1 -> [15,2,2] -> flatten (c*4 + y*2 + x) ----
    for (int i = tid; i < IMGS_PER_BLOCK * 60; i += THREADS) {
        const int img = i / 60, r = i % 60;
        const int oc = r / 4, p = r % 4;
        const int oy = p / 2, ox = p % 2;
        const float* c = C2 + img * 135 + oc * 9;
        FLAT[i] = fmaxf(fmaxf(c[oy * 3 + ox],       c[oy * 3 + ox + 1]),
                        fmaxf(c[(oy + 1) * 3 + ox], c[(oy + 1) * 3 + ox + 1]));
    }
    __syncthreads();

    // ---- Stage 7 (wave 0 only; EXEC all-ones for WMMA): dense tail ----
    if (tid < 32) {
        const int lane = tid;
        const int half = lane >> 4;   // K-pair / row-half selector
        const int nl   = lane & 15;   // A-row (M) or B/D-column (N)

#if __has_builtin(__builtin_amdgcn_wmma_f32_16x16x4_f32)
        // z[16,4] = FLAT[16,60] @ wp^T[60,4]  via 15 chained K=4 f32 WMMAs.
        // A layout: lane holds row m=nl, elements K = 2*half, 2*half+1.
        // B layout (mirrored): lane holds col n=nl, rows K = 2*half, 2*half+1.
        v8f acc = {0.f, 0.f, 0.f, 0.f, 0.f, 0.f, 0.f, 0.f};
#pragma unroll
        for (int k0 = 0; k0 < 60; k0 += 4) {
            v2f a, b;
            a.x = FLAT[nl * 60 + k0 + 2 * half];
            a.y = FLAT[nl * 60 + k0 + 2 * half + 1];
            b.x = WPB[nl * 60 + k0 + 2 * half];       // unconditional: padded panel
            b.y = WPB[nl * 60 + k0 + 2 * half + 1];
            acc = __builtin_amdgcn_wmma_f32_16x16x4_f32(
                false, a, false, b, (short)0, acc, false, false);
        }
        // D layout: lane holds col n=nl, rows m = v + 8*half.
        if (nl < 4) {
#pragma unroll
            for (int v = 0; v < 8; ++v)
                Z[(v + 8 * half) * 4 + nl] = acc[v] + bp[nl];
        }
        if (lane < 16) {
            float z0 = Z[lane * 4 + 0], z1 = Z[lane * 4 + 1];
            float z2 = Z[lane * 4 + 2], z3 = Z[lane * 4 + 3];
            ZZ[lane] = z0 * z0 + z1 * z1 + z2 * z2 + z3 * z3;
        }

        // RBF: ||z-s||^2 = ||z||^2 + ||s||^2 - 2*z.s ; z.s via K=4 f32 WMMA.
        v2f za;
        za.x = Z[nl * 4 + 2 * half];
        za.y = Z[nl * 4 + 2 * half + 1];
        float logit[8];
#pragma unroll
        for (int v = 0; v < 8; ++v) logit[v] = 0.f;

#pragma unroll
        for (int t = 0; t < 4; ++t) {             // 4 tiles of 16 support vecs
            const int n = t * 16 + nl;
            v2f sb;
            sb.x = support[n * 4 + 2 * half];
            sb.y = support[n * 4 + 2 * half + 1];
            v8f d = {0.f, 0.f, 0.f, 0.f, 0.f, 0.f, 0.f, 0.f};
            d = __builtin_amdgcn_wmma_f32_16x16x4_f32(
                false, za, false, sb, (short)0, d, false, false);
            const float s0 = support[n * 4 + 0], s1 = support[n * 4 + 1];
            const float s2 = support[n * 4 + 2], s3 = support[n * 4 + 3];
            const float ss  = s0 * s0 + s1 * s1 + s2 * s2 + s3 * s3;
            const float wln = wl[n];
#pragma unroll
            for (int v = 0; v < 8; ++v) {
                const int m = v + 8 * half;
                const float dist2 = ZZ[m] + ss - 2.0f * d[v];
                logit[v] += expf(-dist2) * wln;   // GAMMA = 1.0
            }
        }
        // Sum the 16 support-columns held across lanes sharing a row-half.
#pragma unroll
        for (int v = 0; v < 8; ++v) {
            float s = logit[v];
            s += __shfl_xor(s, 1, 32);
            s += __shfl_xor(s, 2, 32);
            s += __shfl_xor(s, 4, 32);
            s += __shfl_xor(s, 8, 32);
            logit[v] = s;
        }
        if (nl == 0) {
            const float bias = bl[0];
#pragma unroll
            for (int v = 0; v < 8; ++v) {
                const int m = v + 8 * half;
                const int gimg = img0 + m;
                if (gimg < B) {
                    const float p = 1.0f / (1.0f + expf(-(logit[v] + bias)));
                    out[gimg * 2 + 0] = p;
                    out[gimg * 2 + 1] = 1.0f - p;
                }
            }
        }
#else
        // Scalar fallback (keeps compile green if the builtin is absent).
        if (lane < 16) {
            const int m = lane;
            float z[4];
            for (int c = 0; c < 4; ++c) {
                float a = bp[c];
                for (int k = 0; k < 60; ++k) a += FLAT[m * 60 + k] * wp[c * 60 + k];
                z[c] = a;
            }
            float logit = bl[0];
            for (int n = 0; n < 64; ++n) {
                float d2 = 0.f;
                for (int c = 0; c < 4; ++c) {
                    const float df = z[c] - support[n * 4 + c];
                    d2 += df * df;
                }
                logit += expf(-d2) * wl[n];
            }
            const int gimg = img0 + m;
            if (gimg < B) {
                const float p = 1.0f / (1.0f + expf(-logit));
                out[gimg * 2 + 0] = p;
                out[gimg * 2 + 1] = 1.0f - p;
            }
        }
#endif
    }
}

extern "C" void kernel_launch(void* const* d_in, const int* in_sizes, int n_in,
                              void* d_out, int out_size, void* d_ws, size_t ws_size,
                              hipStream_t stream) {
    const float* x       = (const float*)d_in[0];
    const float* wq      = (const float*)d_in[1];
    const float* bq      = (const float*)d_in[2];
    const float* w1      = (const float*)d_in[3];
    const float* b1      = (const float*)d_in[4];
    const float* w2      = (const float*)d_in[5];
    const float* b2      = (const float*)d_in[6];
    const float* wp      = (const float*)d_in[7];
    const float* bp      = (const float*)d_in[8];
    const float* support = (const float*)d_in[9];
    const float* wl      = (const float*)d_in[10];
    const float* bl      = (const float*)d_in[11];
    float* out = (float*)d_out;

    const int B = in_sizes[0] / (3 * 28 * 28);
    const int blocks = (B + IMGS_PER_BLOCK - 1) / IMGS_PER_BLOCK;

    hqbc_fused_kernel<<<blocks, THREADS, 0, stream>>>(
        x, wq, bq, w1, b1, w2, b2, wp, bp, support, wl, bl, out, B);
}